// WeightedProcrustes_9483287789595
// MI455X (gfx1250) — compile-verified
//
#include <hip/hip_runtime.h>

typedef __attribute__((ext_vector_type(2))) float v2f;
typedef __attribute__((ext_vector_type(8))) float v8f;

#define WAVES_PER_BLOCK 8

__device__ __forceinline__ void swapf(float& a, float& b) { float t = a; a = b; b = t; }
__device__ __forceinline__ void swap3(float* a, float* b) {
    swapf(a[0], b[0]); swapf(a[1], b[1]); swapf(a[2], b[2]);
}

// Solve one batch from the 4x4 raw moment matrix S, write 4x4 transform.
__device__ void solve_procrustes(const float S[4][4], float* __restrict__ o)
{
    const float EPS = 1e-5f;
    const float sw  = S[3][3];
    const float inv = 1.0f / (sw + EPS);
    const float W   = sw * inv;

    float sc[3], rc[3];
    #pragma unroll
    for (int i = 0; i < 3; ++i) { sc[i] = S[i][3] * inv; rc[i] = S[3][i] * inv; }

    // H = (Sigma w s r^T)*inv - (2 - W) * src_c ref_c^T   (centered covariance)
    float H[3][3];
    const float k2 = 2.0f - W;
    #pragma unroll
    for (int i = 0; i < 3; ++i)
        #pragma unroll
        for (int j = 0; j < 3; ++j)
            H[i][j] = S[i][j] * inv - k2 * sc[i] * rc[j];

    // G = H^T H (symmetric PSD)
    float G[3][3];
    #pragma unroll
    for (int i = 0; i < 3; ++i)
        #pragma unroll
        for (int j = 0; j < 3; ++j)
            G[i][j] = H[0][i]*H[0][j] + H[1][i]*H[1][j] + H[2][i]*H[2][j];

    // Cyclic Jacobi eigensolver, 6 sweeps of (0,1),(0,2),(1,2)
    float V[3][3] = {{1,0,0},{0,1,0},{0,0,1}};
    #pragma unroll 1
    for (int sweep = 0; sweep < 6; ++sweep) {
        #pragma unroll
        for (int pair = 0; pair < 3; ++pair) {
            const int p = (pair == 2) ? 1 : 0;
            const int q = (pair == 0) ? 1 : 2;
            const int k = 3 - p - q;
            float apq = G[p][q];
            if (fabsf(apq) > 1e-30f) {
                float th = (G[q][q] - G[p][p]) / (2.0f * apq);
                float tt = copysignf(1.0f, th) / (fabsf(th) + sqrtf(th*th + 1.0f));
                float ct = 1.0f / sqrtf(tt*tt + 1.0f);
                float st = tt * ct;
                G[p][p] -= tt * apq;
                G[q][q] += tt * apq;
                G[p][q] = 0.0f; G[q][p] = 0.0f;
                float gkp = G[k][p], gkq = G[k][q];
                float nkp = ct*gkp - st*gkq;
                float nkq = st*gkp + ct*gkq;
                G[k][p] = nkp; G[p][k] = nkp;
                G[k][q] = nkq; G[q][k] = nkq;
                #pragma unroll
                for (int m = 0; m < 3; ++m) {
                    float vp = V[m][p], vq = V[m][q];
                    V[m][p] = ct*vp - st*vq;
                    V[m][q] = st*vp + ct*vq;
                }
            }
        }
    }

    // Sort eigenpairs descending; columns of V are right singular vectors of H.
    float e0 = G[0][0], e1 = G[1][1], e2 = G[2][2];
    float c0[3] = {V[0][0], V[1][0], V[2][0]};
    float c1[3] = {V[0][1], V[1][1], V[2][1]};
    float c2[3] = {V[0][2], V[1][2], V[2][2]};
    if (e0 < e1) { swapf(e0, e1); swap3(c0, c1); }
    if (e0 < e2) { swapf(e0, e2); swap3(c0, c2); }
    if (e1 < e2) { swapf(e1, e2); swap3(c1, c2); }

    // u_i = H v_i / sigma_i  (normalize + Gram-Schmidt for robustness)
    float u0[3], u1[3], u2[3];
    #pragma unroll
    for (int i = 0; i < 3; ++i) u0[i] = H[i][0]*c0[0] + H[i][1]*c0[1] + H[i][2]*c0[2];
    float n0 = sqrtf(u0[0]*u0[0] + u0[1]*u0[1] + u0[2]*u0[2]);
    float in0 = 1.0f / fmaxf(n0, 1e-30f);
    #pragma unroll
    for (int i = 0; i < 3; ++i) u0[i] *= in0;

    #pragma unroll
    for (int i = 0; i < 3; ++i) u1[i] = H[i][0]*c1[0] + H[i][1]*c1[1] + H[i][2]*c1[2];
    float d01 = u0[0]*u1[0] + u0[1]*u1[1] + u0[2]*u1[2];
    #pragma unroll
    for (int i = 0; i < 3; ++i) u1[i] -= d01 * u0[i];
    float n1 = sqrtf(u1[0]*u1[0] + u1[1]*u1[1] + u1[2]*u1[2]);
    float in1 = 1.0f / fmaxf(n1, 1e-30f);
    #pragma unroll
    for (int i = 0; i < 3; ++i) u1[i] *= in1;

    // u2 = u0 x u1  => det(U) = +1, so sign(det(V U^T)) = sign(det V)
    u2[0] = u0[1]*u1[2] - u0[2]*u1[1];
    u2[1] = u0[2]*u1[0] - u0[0]*u1[2];
    u2[2] = u0[0]*u1[1] - u0[1]*u1[0];

    float detV = c0[0]*(c1[1]*c2[2] - c1[2]*c2[1])
               - c0[1]*(c1[0]*c2[2] - c1[2]*c2[0])
               + c0[2]*(c1[0]*c2[1] - c1[1]*c2[0]);
    float s2 = (detV > 0.0f) ? 1.0f : ((detV < 0.0f) ? -1.0f : 0.0f);

    // R = v0 u0^T + v1 u1^T + s2 * v2 u2^T  ==  V diag(1,1,s) U^T
    float R[3][3];
    #pragma unroll
    for (int i = 0; i < 3; ++i)
        #pragma unroll
        for (int j = 0; j < 3; ++j)
            R[i][j] = c0[i]*u0[j] + c1[i]*u1[j] + s2 * c2[i]*u2[j];

    #pragma unroll
    for (int i = 0; i < 3; ++i) {
        float ti = rc[i] - (R[i][0]*sc[0] + R[i][1]*sc[1] + R[i][2]*sc[2]);
        o[i*4 + 0] = R[i][0]; o[i*4 + 1] = R[i][1]; o[i*4 + 2] = R[i][2];
        o[i*4 + 3] = ti;
    }
    o[12] = 0.0f; o[13] = 0.0f; o[14] = 0.0f; o[15] = 1.0f;
}

// One wave per batch. The 4xN * Nx4 moment GEMM is tiled onto
// V_WMMA_F32_16X16X4_F32: row m = c + 4*ra, col n = d + 4*rb; the four
// diagonal 4x4 blocks of C accumulate S over point phases r = 0..3.
__global__ __launch_bounds__(WAVES_PER_BLOCK * 32)
void wproc_kernel(const float* __restrict__ src, const float* __restrict__ tgt,
                  const float* __restrict__ wgt, float* __restrict__ out,
                  int N, int Btot)
{
    __shared__ float lds[WAVES_PER_BLOCK * 256];

    const int lane  = threadIdx.x & 31;
    const int wv    = threadIdx.x >> 5;
    const int batch = blockIdx.x * WAVES_PER_BLOCK + wv;
    const bool active = batch < Btot;

    const int hi  = lane >> 4;     // which K half (A: K=0,1 vs K=2,3)
    const int L   = lane & 15;
    const int r   = L >> 2;        // point phase within a group of 4
    const int c   = L & 3;         // component: 0..2 = x/y/z, 3 = weight/one
    const int cc  = (c < 3) ? c : 0;
    const bool isw = (c == 3);

    const long base = (long)(active ? batch : 0) * (long)N;
    const int  pl   = 8*hi + r;    // lane's first point within a 16-point tile

    v8f accA = {0,0,0,0,0,0,0,0};
    v8f accB = {0,0,0,0,0,0,0,0};

    const int ntiles = active ? (N >> 4) : 0;

    int t = 0;
    for (; t + 1 < ntiles; t += 2) {
        {
            const long g0 = base + (long)(t << 4) + pl;
            const long g1 = g0 + 4;
            float w0 = wgt[g0], w1 = wgt[g1];
            float s0 = src[g0*3 + cc], s1 = src[g1*3 + cc];
            float r0 = tgt[g0*3 + cc], r1 = tgt[g1*3 + cc];
            v2f a, b;
            a.x = isw ? w0 : w0 * s0;  a.y = isw ? w1 : w1 * s1;
            b.x = isw ? 1.0f : r0;     b.y = isw ? 1.0f : r1;
            accA = __builtin_amdgcn_wmma_f32_16x16x4_f32(
                false, a, false, b, (short)0, accA, false, false);
            __builtin_prefetch(&src[(g0 + 128) * 3], 0, 1);
        }
        {
            const long g0 = base + (long)((t + 1) << 4) + pl;
            const long g1 = g0 + 4;
            float w0 = wgt[g0], w1 = wgt[g1];
            float s0 = src[g0*3 + cc], s1 = src[g1*3 + cc];
            float r0 = tgt[g0*3 + cc], r1 = tgt[g1*3 + cc];
            v2f a, b;
            a.x = isw ? w0 : w0 * s0;  a.y = isw ? w1 : w1 * s1;
            b.x = isw ? 1.0f : r0;     b.y = isw ? 1.0f : r1;
            accB = __builtin_amdgcn_wmma_f32_16x16x4_f32(
                false, a, false, b, (short)0, accB, false, false);
            __builtin_prefetch(&tgt[(g0 + 128) * 3], 0, 1);
        }
    }
    if (t < ntiles) {  // tail (N % 32 == 16 case)
        const long g0 = base + (long)(t << 4) + pl;
        const long g1 = g0 + 4;
        float w0 = wgt[g0], w1 = wgt[g1];
        float s0 = src[g0*3 + cc], s1 = src[g1*3 + cc];
        float r0 = tgt[g0*3 + cc], r1 = tgt[g1*3 + cc];
        v2f a, b;
        a.x = isw ? w0 : w0 * s0;  a.y = isw ? w1 : w1 * s1;
        b.x = isw ? 1.0f : r0;     b.y = isw ? 1.0f : r1;
        accA = __builtin_amdgcn_wmma_f32_16x16x4_f32(
            false, a, false, b, (short)0, accA, false, false);
    }

    v8f acc = accA + accB;

    // Spill C fragment to LDS: VGPR v, lane l holds C[v + 8*hi][l & 15].
    float* ws = &lds[wv * 256];
    #pragma unroll
    for (int v = 0; v < 8; ++v)
        ws[(v + 8*hi) * 16 + L] = acc[v];

    __syncthreads();

    if (active && lane == 0) {
        // Sum the four diagonal 4x4 blocks -> raw moment matrix S.
        float S[4][4];
        #pragma unroll
        for (int i = 0; i < 4; ++i)
            #pragma unroll
            for (int j = 0; j < 4; ++j) {
                float s = 0.0f;
                #pragma unroll
                for (int rr = 0; rr < 4; ++rr)
                    s += ws[(i + 4*rr) * 16 + (j + 4*rr)];
                S[i][j] = s;
            }
        solve_procrustes(S, out + (long)batch * 16);
    }
}

extern "C" void kernel_launch(void* const* d_in, const int* in_sizes, int n_in,
                              void* d_out, int out_size, void* d_ws, size_t ws_size,
                              hipStream_t stream) {
    const float* src = (const float*)d_in[0];   // (B, N, 3) f32
    const float* tgt = (const float*)d_in[1];   // (B, N, 3) f32
    const float* wgt = (const float*)d_in[2];   // (B, N)    f32
    float* out = (float*)d_out;                 // (B, 4, 4) f32

    const int B = out_size / 16;
    const int N = in_sizes[2] / B;

    const int blocks = (B + WAVES_PER_BLOCK - 1) / WAVES_PER_BLOCK;
    wproc_kernel<<<blocks, WAVES_PER_BLOCK * 32, 0, stream>>>(src, tgt, wgt, out, N, B);
}